// LatentDT_10015863734783
// MI455X (gfx1250) — compile-verified
//
#include <hip/hip_runtime.h>

typedef __attribute__((ext_vector_type(2))) float v2f;
typedef __attribute__((ext_vector_type(8))) float v8f;

#define DEPTH     10
#define NBSPLIT   1023      // internal (split) nodes
#define NBNODES   2047      // all nodes
#define KDIM      128
#define ROWS      16        // x rows per workgroup
#define NT        256       // threads per workgroup (8 waves)
#define SX_STRIDE 132       // padded stride for x tile  (conflict-free frag reads)
#define XA_STRIDE 1028      // padded stride for XA tile (conflict-free acc stores)
#define M_STRIDE  1024      // running-min array stride

__global__ __launch_bounds__(NT, 2)
void latentdt_fused(const float* __restrict__ x,
                    const float* __restrict__ A,
                    float* __restrict__ z)
{
    __shared__ alignas(16) float sXA[ROWS * XA_STRIDE]; // 65792 B : XA = x.A^T tile
    __shared__ alignas(16) float sM [ROWS * M_STRIDE];  // 65536 B : running mins
    float* sX = sM;  // x staging tile (16*132 = 2112 floats) aliases sM: dead after GEMM

    const int tid  = threadIdx.x;
    const int wave = tid >> 5;
    const int lane = tid & 31;
    const int half = lane >> 4;   // K-half selector of the WMMA fragment layout
    const int l16  = lane & 15;   // M (for A-frag) / N (for B-frag) index
    const int row0 = blockIdx.x * ROWS;

    // ---- stage x tile (16 x 128 f32) into LDS, float4 per load ----
    for (int i = tid; i < (ROWS * KDIM) / 4; i += NT) {
        const int r = i >> 5;            // / (KDIM/4)
        const int c = (i & 31) << 2;
        const float4 v = *(const float4*)(x + (size_t)(row0 + r) * KDIM + c);
        *(float4*)(&sX[r * SX_STRIDE + c]) = v;
    }
    __syncthreads();

    // ---- per-wave A-fragments for all 32 K-steps (16x4 f32 each, 2 VGPRs) ----
    // lane<16 holds (M=l16, K=k..k+1) ; lane>=16 holds (M=l16, K=k+2..k+3)
    v2f af[32];
    #pragma unroll
    for (int ks = 0; ks < 32; ++ks)
        af[ks] = *(const v2f*)(&sX[l16 * SX_STRIDE + (ks << 2) + (half << 1)]);
    __syncthreads();   // sX region is reused as sM below

    // ---- GEMM: XA[16 x 1023] via V_WMMA_F32_16X16X4_F32, waves split 64 col-tiles ----
    for (int t = wave; t < 64; t += 8) {
        const int col0 = t << 4;
        int node = col0 + l16;
        if (node > NBSPLIT - 1) node = NBSPLIT - 1;        // pad column 1023 safely
        const float* Ap = A + (size_t)node * KDIM + (half << 1);
        __builtin_prefetch(Ap + (size_t)128 * KDIM, 0, 0); // hint next tile of A
        v8f acc = {};
        #pragma unroll
        for (int ks = 0; ks < 32; ++ks) {
            const v2f b = *(const v2f*)(Ap + (ks << 2));   // B = A^T fragment
            acc = __builtin_amdgcn_wmma_f32_16x16x4_f32(
                false, af[ks], false, b, (short)0, acc, false, false);
        }
        // C layout: lane half selects M in [0,8) or [8,16); N = l16
        const int srow = half << 3;
        #pragma unroll
        for (int r = 0; r < 8; ++r)
            sXA[(srow + r) * XA_STRIDE + col0 + l16] = acc[r];
    }
    __syncthreads();

    // ---- root node: m(0) = 1, z(row, 0) = 1 ----
    if (tid < ROWS) {
        sM[tid * M_STRIDE] = 1.0f;
        __builtin_nontemporal_store(1.0f, z + (size_t)(row0 + tid) * NBNODES);
    }
    __syncthreads();

    // ---- tree propagation, level by level ----
    // m(2p+1) = min(m(p), XA_p) ; m(2p+2) = min(m(p), -XA_p) ; z = clamp(m, 0, 1)
    int lvlBase = 0;
    for (int d = 0; d < DEPTH; ++d) {
        const int lvlSize = 1 << d;
        const int tasks   = ROWS << d;
        for (int i = tid; i < tasks; i += NT) {
            const int r = i >> d;
            const int p = lvlBase + (i & (lvlSize - 1));
            const float mp = sM[r * M_STRIDE + p];
            const float xa = sXA[r * XA_STRIDE + p];
            const float mL = fminf(mp,  xa);
            const float mR = fminf(mp, -xa);
            const int cL = 2 * p + 1;
            if (d < DEPTH - 1) {                 // children are split nodes too
                sM[r * M_STRIDE + cL]     = mL;
                sM[r * M_STRIDE + cL + 1] = mR;
            }
            float* zp = z + (size_t)(row0 + r) * NBNODES + cL;
            __builtin_nontemporal_store(fminf(fmaxf(mL, 0.0f), 1.0f), zp);
            __builtin_nontemporal_store(fminf(fmaxf(mR, 0.0f), 1.0f), zp + 1);
        }
        lvlBase += lvlSize;
        __syncthreads();
    }
}

extern "C" void kernel_launch(void* const* d_in, const int* in_sizes, int n_in,
                              void* d_out, int out_size, void* d_ws, size_t ws_size,
                              hipStream_t stream)
{
    (void)n_in; (void)out_size; (void)d_ws; (void)ws_size;
    const float* x = (const float*)d_in[0];   // [32768, 128] f32
    const float* A = (const float*)d_in[1];   // [1023, 128]  f32
    float* zout    = (float*)d_out;           // [32768, 2047] f32

    const int N = in_sizes[0] / KDIM;         // 32768
    dim3 grid(N / ROWS), block(NT);
    latentdt_fused<<<grid, block, 0, stream>>>(x, A, zout);
}